// GARCHGen_46608985096245
// MI455X (gfx1250) — compile-verified
//
#include <hip/hip_runtime.h>
#include <stdint.h>

// GARCH(1,1) path generation, N=32768 paths, T=2048 steps, D=2 dims.
// Memory-bound scan (~1.07 GB traffic -> ~46us floor at 23.3 TB/s).
// CDNA5 path: triple-buffered async global->LDS DMA (ASYNCcnt), wave-private
// LDS slices (zero barriers), sequential recurrence in registers, coalesced
// b128 write-out. sqrt kept out of the carried state so v_sqrt_f32 is safe.

#define T_STEPS 2048
#define DIMS 2
#define TB 64                   // timesteps per tile
#define NT (T_STEPS / TB)       // 32 tiles
#define ROWS_W 16               // paths per wave (2 threads per path)
#define WAVES 2                 // waves per block
#define ROWS_B (ROWS_W * WAVES) // 32 paths per block
#define NBUF 3                  // triple buffer -> prefetch depth 2
#define LROW_F 132              // floats per LDS row: 128 data + 4 pad = 528 B
#define ROW_BYTES ((size_t)T_STEPS * DIMS * 4)  // 16384 B per path
#define TILE_BYTES (TB * DIMS * 4)              // 512 B per row per tile

__global__ __launch_bounds__(64) void garch_scan_kernel(
    const float* __restrict__ z,
    const float* __restrict__ omega_p, const float* __restrict__ alpha_p,
    const float* __restrict__ beta_p,  const float* __restrict__ mu_p,
    const float* __restrict__ s20_p,
    float* __restrict__ x)
{
  // [3 buffers][wave][16 rows * 132 floats] = 50688 B
  __shared__ float tile[NBUF][WAVES][ROWS_W * LROW_F];

  const int tid  = threadIdx.x;
  const int wave = tid >> 5;
  const int lane = tid & 31;
  const int pw0  = blockIdx.x * ROWS_B + wave * ROWS_W; // first path of this wave
  const int d    = lane & 1;                            // dimension handled
  const int myr  = lane >> 1;                           // my path row in slice

  // per-thread GARCH parameters (D==2, d = lane&1)
  const float omega = omega_p[d];
  const float alpha = alpha_p[d];
  const float beta  = beta_p[d];
  const float mu    = mu_p[d];
  float sig2 = s20_p[d];   // carried clamped variance
  float eps2 = 0.0f;       // carried eps^2 (exact: s * z^2, no sqrt involved)

  const char* zb = (const char*)z;
  const size_t wrow0 = (size_t)pw0 * ROW_BYTES;

  // One async b128 instruction = 32 lanes * 16B = one full 512B row segment.
  // Issue 16 per tile (one per row); ASYNCcnt +16 per tile batch.
#define ISSUE_TILE_LOADS(buf, kk)                                              \
  {                                                                            \
    float* nslice = &tile[(buf)][wave][0];                                     \
    const size_t tofs = (size_t)(kk) * TILE_BYTES + (size_t)lane * 16;         \
    _Pragma("unroll")                                                          \
    for (int i = 0; i < ROWS_W; ++i) {                                         \
      const unsigned ldsb =                                                    \
          (unsigned)(uintptr_t)(&nslice[i * LROW_F]) + (unsigned)(lane * 16);  \
      const char* g = zb + wrow0 + (size_t)i * ROW_BYTES + tofs;               \
      asm volatile("global_load_async_to_lds_b128 %0, %1, off"                 \
                   :: "v"(ldsb), "v"(g) : "memory");                           \
    }                                                                          \
  }

  // ---- prefetch tiles 0 and 1 ----
  ISSUE_TILE_LOADS(0, 0)
  ISSUE_TILE_LOADS(1, 1)

  for (int k = 0; k < NT; ++k) {
    const int cur = k % NBUF;
    float* slice = &tile[cur][wave][0];

    if (k + 2 < NT) {
      // buffer (k+2)%3 held tile k-1, whose LDS readback retired during
      // iteration k-1 (DScnt waits before its global stores).
      ISSUE_TILE_LOADS((k + 2) % NBUF, k + 2)
      // all async ops are loads -> in-order completion: <=32 outstanding
      // means tile k's 16 are done (only k+1, k+2 batches may remain).
      asm volatile("s_wait_asynccnt 0x20" ::: "memory");
    } else if (k + 1 < NT) {
      asm volatile("s_wait_asynccnt 0x10" ::: "memory");
    } else {
      asm volatile("s_wait_asynccnt 0x0" ::: "memory");
    }

    // ---- sequential GARCH recurrence, in-place in LDS (wave-private) ----
    // bank: dword = (lane>>1)*132 + 2t + (lane&1) -> 4*(lane>>1)+(lane&1)+2t,
    // all 32 lanes distinct mod 64 -> conflict-free.
    float* row = &slice[myr * LROW_F];
#pragma unroll
    for (int t = 0; t < TB; ++t) {
      const float zt = row[2 * t + d];
      float s = omega + alpha * eps2 + beta * sig2;
      s = fmaxf(s, 1e-12f);
      // carried state is sqrt-free and exact: eps^2 = s * z^2
      eps2 = s * (zt * zt);
      sig2 = s;
      // output only: single v_sqrt_f32, error does not feed the recurrence
      row[2 * t + d] = mu + __builtin_amdgcn_sqrtf(s) * zt;
    }

    // ---- coalesced write-out: b128 LDS readback -> 512B contiguous stores.
    const size_t tofs = (size_t)k * TILE_BYTES + (size_t)lane * 16;
#pragma unroll
    for (int i = 0; i < ROWS_W; ++i) {
      const float4 v = *(const float4*)(&slice[i * LROW_F + lane * 4]);
      *(float4*)((char*)x + wrow0 + (size_t)i * ROW_BYTES + tofs) = v;
    }
  }
#undef ISSUE_TILE_LOADS
}

extern "C" void kernel_launch(void* const* d_in, const int* in_sizes, int n_in,
                              void* d_out, int out_size, void* d_ws, size_t ws_size,
                              hipStream_t stream)
{
  const float* z     = (const float*)d_in[0];
  const float* omega = (const float*)d_in[1];
  const float* alpha = (const float*)d_in[2];
  const float* beta  = (const float*)d_in[3];
  const float* mu    = (const float*)d_in[4];
  const float* s20   = (const float*)d_in[5];
  float* x = (float*)d_out;

  const int N = in_sizes[0] / (T_STEPS * DIMS);   // 32768
  dim3 grid((unsigned)(N / ROWS_B));              // 1024 blocks
  dim3 block(WAVES * 32);                         // 64 threads = 2 waves
  garch_scan_kernel<<<grid, block, 0, stream>>>(z, omega, alpha, beta, mu, s20, x);
}